// ZBL_3573412790692
// MI455X (gfx1250) — compile-verified
//
#include <hip/hip_runtime.h>
#include <math.h>

// ---------------------------------------------------------------------------
// ZBL screened-repulsion edge kernel for MI455X (gfx1250, wave32).
// Memory-bound problem: 192 MB edge stream @ 23.3 TB/s ~= 8.3 us floor.
// Strategy: precompute Z^p (per atom in ws, or 96-entry LDS table staged with
// gfx1250 async-to-LDS), fuse exp constants into exp2 form (native v_exp_f32),
// 128-bit non-temporal streaming loads, run-length-compressed segment sum.
// ---------------------------------------------------------------------------

#define LOG2E_F 1.4426950408889634f
#define ZTAB_N 96                 // padded 94 -> 96 (24 x 16B async b128 loads)
#define EPT 16                    // edges per thread
#define BLOCK 256                 // 8 wave32 waves

typedef float __attribute__((ext_vector_type(4))) f32x4;
typedef int   __attribute__((ext_vector_type(4))) i32x4;

#if defined(__has_builtin)
#  if __has_builtin(__builtin_amdgcn_exp2f)
#    define FAST_EXP2(x) __builtin_amdgcn_exp2f(x)
#  else
#    define FAST_EXP2(x) exp2f(x)
#  endif
#else
#  define FAST_EXP2(x) exp2f(x)
#endif

// --- setup: fused constants  e2[k] = -a[k]*log2(e)/d ; copy c[k] ------------
__global__ void zbl_setup_consts(const float* __restrict__ dptr,
                                 const float* __restrict__ c,
                                 const float* __restrict__ a,
                                 float* __restrict__ hdr) {
  int k = threadIdx.x;
  if (k < 4) {
    float dinv = 1.0f / dptr[0];
    hdr[k]     = -a[k] * LOG2E_F * dinv;
    hdr[4 + k] = c[k];
  }
}

// --- setup: per-atom Z^p lookup (preferred path, 2 MB in ws, L2-resident) ---
__global__ void zbl_zpow_atoms(const int* __restrict__ Z,
                               const float* __restrict__ pptr,
                               float* __restrict__ zp, int n) {
  int i = blockIdx.x * blockDim.x + threadIdx.x;
  if (i < n) zp[i] = powf((float)Z[i], pptr[0]);
}

// --- setup: per-Z Z^p table (fallback if ws too small) ----------------------
__global__ void zbl_zpow_table(const float* __restrict__ pptr,
                               float* __restrict__ tab) {
  int z = threadIdx.x;
  if (z < ZTAB_N) tab[z] = powf((float)z, pptr[0]);
}

// --- main edge kernel -------------------------------------------------------
template <bool ATOM_LUT>
__global__ __launch_bounds__(BLOCK) void zbl_edge_kernel(
    const float* __restrict__ dij,
    const int*   __restrict__ idx_i,
    const int*   __restrict__ idx_j,
    const int*   __restrict__ Z,     // only used when !ATOM_LUT
    const float* __restrict__ lut,   // per-atom Z^p  or  96-entry Z^p table
    const float* __restrict__ hdr,   // e2[0..3], c[0..3]
    float* __restrict__ out,
    int E) {
  __shared__ float s_lut[ZTAB_N];

  if constexpr (!ATOM_LUT) {
    // Stage the 96-entry table into LDS with the CDNA5 async tensor-class
    // copy path (GLOBAL_LOAD_ASYNC_TO_LDS_B128, tracked by ASYNCcnt).
    if (threadIdx.x < (ZTAB_N * 4 / 16)) {
      unsigned lds_addr = (unsigned)(unsigned long long)&s_lut[0]
                        + threadIdx.x * 16u;
      unsigned voff = threadIdx.x * 16u;
      unsigned long long gbase = (unsigned long long)(const void*)lut;
      asm volatile("global_load_async_to_lds_b128 %0, %1, %2 offset:0"
                   :
                   : "v"(lds_addr), "v"(voff), "s"(gbase)
                   : "memory");
    }
    asm volatile("s_wait_asynccnt 0" ::: "memory");
    __syncthreads();
  }

  float e0 = hdr[0], e1 = hdr[1], e2 = hdr[2], e3 = hdr[3];
  float c0 = hdr[4], c1 = hdr[5], c2 = hdr[6], c3 = hdr[7];

  int tid  = blockIdx.x * blockDim.x + threadIdx.x;
  int base = tid * EPT;
  if (base >= E) return;
  int n = (E - base < EPT) ? (E - base) : EPT;

  int   cur = -1;
  float acc = 0.0f;

  // Per-edge body: gather Z^p, fused exp2 screening sum, run-length segsum.
  auto proc = [&](float dv, int ai, int aj) {
    float zpi, zpj;
    if constexpr (ATOM_LUT) {
      zpi = lut[ai];
      zpj = lut[aj];
    } else {
      zpi = s_lut[Z[ai]];
      zpj = s_lut[Z[aj]];
    }
    float t = dv * (zpi + zpj);
    float v = c0 * FAST_EXP2(e0 * t) + c1 * FAST_EXP2(e1 * t)
            + c2 * FAST_EXP2(e2 * t) + c3 * FAST_EXP2(e3 * t);
    if (ai != cur) {
      if (cur >= 0) atomicAdd(out + cur, acc);
      cur = ai;
      acc = v;
    } else {
      acc += v;
    }
  };

  if (n == EPT) {
    // gfx1250 prefetch one block-chunk ahead (speculative, OOB-safe by guard)
    int pf = base + BLOCK * EPT;
    if (pf + EPT <= E) {
      __builtin_prefetch(dij + pf, 0, 0);
      __builtin_prefetch(idx_i + pf, 0, 0);
      __builtin_prefetch(idx_j + pf, 0, 0);
    }
    const f32x4* d4 = (const f32x4*)(dij + base);
    const i32x4* i4 = (const i32x4*)(idx_i + base);
    const i32x4* j4 = (const i32x4*)(idx_j + base);
#pragma unroll
    for (int it = 0; it < EPT / 4; ++it) {
      // stream-once data: non-temporal b128 loads keep L2 for the LUT
      f32x4 dd = __builtin_nontemporal_load(d4 + it);
      i32x4 ii = __builtin_nontemporal_load(i4 + it);
      i32x4 jj = __builtin_nontemporal_load(j4 + it);
      proc(dd.x, ii.x, jj.x);
      proc(dd.y, ii.y, jj.y);
      proc(dd.z, ii.z, jj.z);
      proc(dd.w, ii.w, jj.w);
    }
  } else {
    for (int l = 0; l < n; ++l)
      proc(dij[base + l], idx_i[base + l], idx_j[base + l]);
  }

  if (cur >= 0) atomicAdd(out + cur, acc);
}

// ---------------------------------------------------------------------------
extern "C" void kernel_launch(void* const* d_in, const int* in_sizes, int n_in,
                              void* d_out, int out_size, void* d_ws, size_t ws_size,
                              hipStream_t stream) {
  (void)n_in;
  const int*   Z     = (const int*)d_in[0];
  const float* Dij   = (const float*)d_in[1];
  const int*   idx_i = (const int*)d_in[2];
  const int*   idx_j = (const int*)d_in[3];
  const float* p     = (const float*)d_in[4];
  const float* d     = (const float*)d_in[5];
  const float* c     = (const float*)d_in[6];
  const float* a     = (const float*)d_in[7];
  float* out = (float*)d_out;

  const int nAtoms = in_sizes[0];
  const int E      = in_sizes[1];

  float* hdr = (float*)d_ws;                       // 8 floats (64 B reserved)
  float* lut = (float*)((char*)d_ws + 64);         // Z^p data

  // Output is accumulated atomically -> must be zeroed every call
  // (hipMemsetAsync is graph-capture safe).
  hipMemsetAsync(d_out, 0, (size_t)out_size * sizeof(float), stream);

  zbl_setup_consts<<<1, 32, 0, stream>>>(d, c, a, hdr);

  const size_t needAtom = 64 + (size_t)nAtoms * sizeof(float);
  const bool atomPath = ws_size >= needAtom;
  if (atomPath) {
    zbl_zpow_atoms<<<(nAtoms + 255) / 256, 256, 0, stream>>>(Z, p, lut, nAtoms);
  } else {
    zbl_zpow_table<<<1, 128, 0, stream>>>(p, lut);
  }

  const long long nThreads = ((long long)E + EPT - 1) / EPT;
  const int grid = (int)((nThreads + BLOCK - 1) / BLOCK);
  if (atomPath) {
    zbl_edge_kernel<true><<<grid, BLOCK, 0, stream>>>(
        Dij, idx_i, idx_j, Z, lut, hdr, out, E);
  } else {
    zbl_edge_kernel<false><<<grid, BLOCK, 0, stream>>>(
        Dij, idx_i, idx_j, Z, lut, hdr, out, E);
  }
}